// GAT_18932215840817
// MI455X (gfx1250) — compile-verified
//
#include <hip/hip_runtime.h>
#include <math.h>

// ---------------------------------------------------------------------------
// GAT 3-layer forward for MI455X (gfx1250, wave32).
// GEMMs via V_WMMA_F32_16X16X4_F32 (exact fp32). Edge softmax/aggregation via
// global float atomics; segment-max via monotonic-uint atomicMax.
// ---------------------------------------------------------------------------

typedef float v2f __attribute__((ext_vector_type(2)));
typedef float v8f __attribute__((ext_vector_type(8)));

#define SLOPE 0.2f
#define EPS_BN 1e-5f

__device__ __forceinline__ unsigned f2mono(float f) {
  unsigned u = __float_as_uint(f);
  return (u & 0x80000000u) ? ~u : (u | 0x80000000u);
}
__device__ __forceinline__ float mono2f(unsigned m) {
  unsigned u = (m & 0x80000000u) ? (m ^ 0x80000000u) : ~m;
  return __uint_as_float(u);
}
__device__ __forceinline__ float leaky(float v) { return v > 0.f ? v : SLOPE * v; }

// ---------------------------------------------------------------------------
// GEMM: C[M,NCOLS] = A[M,128] * B[128,NCOLS], all row-major fp32.
// One wave computes one 16x16 tile of C. blockIdx.x = M-tile, wave id = N-tile.
// WMMA f32 16x16x4, 32 steps over K=128. NCOLS is compile-time so all B/C
// row strides become immediate offsets (no per-load 64-bit address math).
// A frag (16x4):  lanes 0-15 hold rows M=0..15 with K={0,1}; lanes 16-31 K={2,3}.
// B frag (4x16):  VGPR0 = K-row (0 + 2*half), VGPR1 = K-row (1 + 2*half); col = lane&15.
// C frag (16x16): VGPR r -> row (r + 8*half), col = lane&15.
// FULL=true: all wave columns valid -> no guards. FULL=false: branchless ragged
// edge (clamped address + cndmask zero, no EXEC change around the WMMA).
// ---------------------------------------------------------------------------
template <int NCOLS, bool FULL>
__global__ void gemm_wmma_f32(const float* __restrict__ A, const float* __restrict__ B,
                              float* __restrict__ C, int M) {
  const int K = 128;
  const int lane = threadIdx.x & 31;
  const int wave = threadIdx.x >> 5;
  const int row0 = blockIdx.x * 16;
  const int col0 = wave * 16;
  const int half = lane >> 4;     // 0: lanes 0-15, 1: lanes 16-31
  const int l    = lane & 15;
  const int kb   = 2 * half;      // K sub-offset for this half-wave

  const int ncol  = col0 + l;
  const bool colok = FULL || (ncol < NCOLS);
  const int ncl   = FULL ? ncol : (colok ? ncol : (NCOLS - 1));  // always-valid col

  const float2* Ap = (const float2*)(A + (size_t)(row0 + l) * K + kb);
  const float*  Bp = B + (size_t)kb * NCOLS + ncl;

  v8f acc = {};
  #pragma unroll 8
  for (int k = 0; k < K; k += 4) {
    // Constant-offset loads: (k/4)*4*NCOLS and +NCOLS fold into offset: fields.
    float2 av = Ap[k >> 1];                  // A[row, k+kb .. k+kb+1]
    float bx = Bp[(size_t)k * NCOLS];        // B[k+kb    , ncl]
    float by = Bp[(size_t)(k + 1) * NCOLS];  // B[k+kb+1  , ncl]
    if (!FULL && !colok) { bx = 0.f; by = 0.f; }  // cndmask, no exec branch
    v2f a; a.x = av.x; a.y = av.y;
    v2f b; b.x = bx;   b.y = by;
    acc = __builtin_amdgcn_wmma_f32_16x16x4_f32(
        /*neg_a=*/false, a, /*neg_b=*/false, b,
        /*c_mod=*/(short)0, acc, /*reuse_a=*/false, /*reuse_b=*/false);
  }
  if (colok) {
    float* Cp = C + (size_t)(row0 + 8 * half) * NCOLS + ncol;
    #pragma unroll
    for (int r = 0; r < 8; ++r)
      Cp[(size_t)r * NCOLS] = acc[r];        // constant offsets: r*NCOLS*4 bytes
  }
}

// ---------------------------------------------------------------------------
// Per-(node,head) attention logits: als[n,h] = <h[n,h,:], a_src[h,:]>, same for ald.
// ---------------------------------------------------------------------------
__global__ void attn_logits(const float* __restrict__ h, const float* __restrict__ a_src,
                            const float* __restrict__ a_dst, float* __restrict__ als,
                            float* __restrict__ ald, int Nn, int H, int Cc) {
  int i = blockIdx.x * blockDim.x + threadIdx.x;
  if (i >= Nn * H) return;
  int n = i / H, hh = i - n * H;
  const float* hv = h + (size_t)n * H * Cc + hh * Cc;
  const float* as = a_src + hh * Cc;
  const float* ad = a_dst + hh * Cc;
  float s = 0.f, d = 0.f;
  for (int c = 0; c < Cc; ++c) { float v = hv[c]; s += v * as[c]; d += v * ad[c]; }
  als[i] = s; ald[i] = d;
}

// ---------------------------------------------------------------------------
// Init per-layer buffers: segmax = mono(-inf), denom = 0, out[n*D+c] = bias[c].
// ---------------------------------------------------------------------------
__global__ void init_layer(unsigned* __restrict__ segmax, float* __restrict__ denom,
                           float* __restrict__ out, const float* __restrict__ bias,
                           int NH, int ND, int D) {
  int i = blockIdx.x * blockDim.x + threadIdx.x;
  if (i < NH) { segmax[i] = f2mono(-INFINITY); denom[i] = 0.f; }
  if (i < ND) out[i] = bias[i % D];
}

__global__ void zero_f32(float* __restrict__ p, int n) {
  int i = blockIdx.x * blockDim.x + threadIdx.x;
  if (i < n) p[i] = 0.f;
}

// ---------------------------------------------------------------------------
// Edge pass 1: segment max of leaky-relu logits over destination.
// One thread per (edge, head). Edges [0,E) from edge_index, [E,E+N) self loops.
// ---------------------------------------------------------------------------
__global__ void edge_logit_max(const int* __restrict__ ei, int E, int Nn, int H,
                               const float* __restrict__ als, const float* __restrict__ ald,
                               unsigned* __restrict__ segmax) {
  int t = blockIdx.x * blockDim.x + threadIdx.x;
  int tot = (E + Nn) * H;
  if (t >= tot) return;
  int e = t / H, hh = t - e * H;
  int s = (e < E) ? ei[e] : (e - E);
  int d = (e < E) ? ei[E + e] : (e - E);
  float v = leaky(als[s * H + hh] + ald[d * H + hh]);
  atomicMax(&segmax[d * H + hh], f2mono(v));
}

// Edge pass 2: denom[d,h] += exp(e - m[d,h])
__global__ void edge_logit_sum(const int* __restrict__ ei, int E, int Nn, int H,
                               const float* __restrict__ als, const float* __restrict__ ald,
                               const unsigned* __restrict__ segmax, float* __restrict__ denom) {
  int t = blockIdx.x * blockDim.x + threadIdx.x;
  int tot = (E + Nn) * H;
  if (t >= tot) return;
  int e = t / H, hh = t - e * H;
  int s = (e < E) ? ei[e] : (e - E);
  int d = (e < E) ? ei[E + e] : (e - E);
  float v = leaky(als[s * H + hh] + ald[d * H + hh]);
  float p = expf(v - mono2f(segmax[d * H + hh]));
  atomicAdd(&denom[d * H + hh], p);
}

// ---------------------------------------------------------------------------
// Edge pass 3 (D=128, H=4, C=32): one wave32 per edge; lane handles channels
// [4*lane, 4*lane+3], head = lane/8. float4 gather of h[src], 4 atomic adds.
// ---------------------------------------------------------------------------
__global__ void edge_agg_d128(const int* __restrict__ ei, int E, int Nn,
                              const float* __restrict__ als, const float* __restrict__ ald,
                              const unsigned* __restrict__ segmax, const float* __restrict__ denom,
                              const float* __restrict__ h, float* __restrict__ out) {
  int t = blockIdx.x * blockDim.x + threadIdx.x;
  int e = t >> 5, lane = t & 31;
  if (e >= E + Nn) return;
  int s = (e < E) ? ei[e] : (e - E);
  int d = (e < E) ? ei[E + e] : (e - E);
  int head = lane >> 3;
  float v = leaky(als[s * 4 + head] + ald[d * 4 + head]);
  float m = mono2f(segmax[d * 4 + head]);
  float alpha = expf(v - m) / (denom[d * 4 + head] + 1e-16f);
  float4 hv = *(const float4*)(h + (size_t)s * 128 + lane * 4);
  float* o = out + (size_t)d * 128 + lane * 4;
  atomicAdd(o + 0, alpha * hv.x);
  atomicAdd(o + 1, alpha * hv.y);
  atomicAdd(o + 2, alpha * hv.z);
  atomicAdd(o + 3, alpha * hv.w);
}

// Edge pass 3 (D=40, H=1): one wave per edge; lane covers channel lane and 32+lane.
__global__ void edge_agg_d40(const int* __restrict__ ei, int E, int Nn,
                             const float* __restrict__ als, const float* __restrict__ ald,
                             const unsigned* __restrict__ segmax, const float* __restrict__ denom,
                             const float* __restrict__ h, float* __restrict__ out) {
  int t = blockIdx.x * blockDim.x + threadIdx.x;
  int e = t >> 5, lane = t & 31;
  if (e >= E + Nn) return;
  int s = (e < E) ? ei[e] : (e - E);
  int d = (e < E) ? ei[E + e] : (e - E);
  float v = leaky(als[s] + ald[d]);
  float alpha = expf(v - mono2f(segmax[d])) / (denom[d] + 1e-16f);
  const float* hs = h + (size_t)s * 40;
  float* o = out + (size_t)d * 40;
  atomicAdd(o + lane, alpha * hs[lane]);
  if (lane < 8) atomicAdd(o + 32 + lane, alpha * hs[32 + lane]);
}

// ---------------------------------------------------------------------------
// BatchNorm stats: 128 channels; thread = channel, blocks stride rows (coalesced).
// ---------------------------------------------------------------------------
__global__ void bn_stats(const float* __restrict__ x, float* __restrict__ sum,
                         float* __restrict__ sumsq, int Nn) {
  int c = threadIdx.x;  // 0..127
  float s = 0.f, ss = 0.f;
  for (int n = blockIdx.x; n < Nn; n += gridDim.x) {
    float v = x[(size_t)n * 128 + c];
    s += v; ss += v * v;
  }
  atomicAdd(&sum[c], s);
  atomicAdd(&sumsq[c], ss);
}

__global__ void bn_apply_elu(float* __restrict__ x, const float* __restrict__ sum,
                             const float* __restrict__ sumsq, const float* __restrict__ gamma,
                             const float* __restrict__ beta, int Nn) {
  int i = blockIdx.x * blockDim.x + threadIdx.x;
  if (i >= Nn * 128) return;
  int c = i & 127;
  float inv = 1.f / (float)Nn;
  float mu = sum[c] * inv;
  float var = sumsq[c] * inv - mu * mu;
  float v = gamma[c] * (x[i] - mu) * rsqrtf(var + EPS_BN) + beta[c];
  x[i] = v > 0.f ? v : expf(v) - 1.f;
}

// ---------------------------------------------------------------------------
// Host orchestration
// ---------------------------------------------------------------------------
static inline int ceil_div(long long a, long long b) { return (int)((a + b - 1) / b); }

extern "C" void kernel_launch(void* const* d_in, const int* in_sizes, int n_in,
                              void* d_out, int out_size, void* d_ws, size_t ws_size,
                              hipStream_t stream) {
  const float* x   = (const float*)d_in[0];
  const int*   ei  = (const int*)d_in[1];
  const float* w1  = (const float*)d_in[2];
  const float* as1 = (const float*)d_in[3];
  const float* ad1 = (const float*)d_in[4];
  const float* b1  = (const float*)d_in[5];
  const float* g1  = (const float*)d_in[6];
  const float* be1 = (const float*)d_in[7];
  const float* w2  = (const float*)d_in[8];
  const float* as2 = (const float*)d_in[9];
  const float* ad2 = (const float*)d_in[10];
  const float* b2  = (const float*)d_in[11];
  const float* g2  = (const float*)d_in[12];
  const float* be2 = (const float*)d_in[13];
  const float* w3  = (const float*)d_in[14];
  const float* as3 = (const float*)d_in[15];
  const float* ad3 = (const float*)d_in[16];
  const float* b3  = (const float*)d_in[17];

  const int N = in_sizes[0] / 128;   // 50000
  const int E = in_sizes[1] / 2;     // 800000
  const int Etot = E + N;            // with self loops
  const int D = 128, H = 4, C = 32, OUTC = 40;

  // Workspace carve-up (256B aligned).
  size_t off = 0;
  auto carve = [&](size_t bytes) -> void* {
    void* p = (char*)d_ws + off;
    off = (off + bytes + 255) & ~(size_t)255;
    return p;
  };
  float*    buf_h   = (float*)carve((size_t)N * D * sizeof(float));   // GEMM output h
  float*    buf_agg = (float*)carve((size_t)N * D * sizeof(float));   // aggregation / next input
  float*    als     = (float*)carve((size_t)N * H * sizeof(float));
  float*    ald     = (float*)carve((size_t)N * H * sizeof(float));
  unsigned* segmax  = (unsigned*)carve((size_t)N * H * sizeof(unsigned));
  float*    denom   = (float*)carve((size_t)N * H * sizeof(float));
  float*    bnsum   = (float*)carve(128 * sizeof(float));
  float*    bnsq    = (float*)carve(128 * sizeof(float));
  (void)ws_size; (void)n_in;

  const int Mtiles = N / 16;  // 3125

  // ---------------- Layer 1 (input = x) ----------------
  gemm_wmma_f32<128, true><<<Mtiles, 256, 0, stream>>>(x, w1, buf_h, N);
  attn_logits<<<ceil_div((long long)N * H, 256), 256, 0, stream>>>(buf_h, as1, ad1, als, ald, N, H, C);
  init_layer<<<ceil_div((long long)N * D, 256), 256, 0, stream>>>(segmax, denom, buf_agg, b1, N * H, N * D, D);
  edge_logit_max<<<ceil_div((long long)Etot * H, 256), 256, 0, stream>>>(ei, E, N, H, als, ald, segmax);
  edge_logit_sum<<<ceil_div((long long)Etot * H, 256), 256, 0, stream>>>(ei, E, N, H, als, ald, segmax, denom);
  edge_agg_d128<<<ceil_div((long long)Etot * 32, 256), 256, 0, stream>>>(ei, E, N, als, ald, segmax, denom, buf_h, buf_agg);
  zero_f32<<<1, 256, 0, stream>>>(bnsum, 128);
  zero_f32<<<1, 256, 0, stream>>>(bnsq, 128);
  bn_stats<<<256, 128, 0, stream>>>(buf_agg, bnsum, bnsq, N);
  bn_apply_elu<<<ceil_div((long long)N * D, 256), 256, 0, stream>>>(buf_agg, bnsum, bnsq, g1, be1, N);

  // ---------------- Layer 2 (input = buf_agg) ----------------
  gemm_wmma_f32<128, true><<<Mtiles, 256, 0, stream>>>(buf_agg, w2, buf_h, N);
  attn_logits<<<ceil_div((long long)N * H, 256), 256, 0, stream>>>(buf_h, as2, ad2, als, ald, N, H, C);
  init_layer<<<ceil_div((long long)N * D, 256), 256, 0, stream>>>(segmax, denom, buf_agg, b2, N * H, N * D, D);
  edge_logit_max<<<ceil_div((long long)Etot * H, 256), 256, 0, stream>>>(ei, E, N, H, als, ald, segmax);
  edge_logit_sum<<<ceil_div((long long)Etot * H, 256), 256, 0, stream>>>(ei, E, N, H, als, ald, segmax, denom);
  edge_agg_d128<<<ceil_div((long long)Etot * 32, 256), 256, 0, stream>>>(ei, E, N, als, ald, segmax, denom, buf_h, buf_agg);
  zero_f32<<<1, 256, 0, stream>>>(bnsum, 128);
  zero_f32<<<1, 256, 0, stream>>>(bnsq, 128);
  bn_stats<<<256, 128, 0, stream>>>(buf_agg, bnsum, bnsq, N);
  bn_apply_elu<<<ceil_div((long long)N * D, 256), 256, 0, stream>>>(buf_agg, bnsum, bnsq, g2, be2, N);

  // ---------------- Layer 3 (H=1, C=40, output -> d_out) ----------------
  float* out = (float*)d_out;
  gemm_wmma_f32<40, false><<<Mtiles, 96, 0, stream>>>(buf_agg, w3, buf_h, N);  // 3 waves/block
  attn_logits<<<ceil_div((long long)N, 256), 256, 0, stream>>>(buf_h, as3, ad3, als, ald, N, 1, OUTC);
  init_layer<<<ceil_div((long long)N * OUTC, 256), 256, 0, stream>>>(segmax, denom, out, b3, N, N * OUTC, OUTC);
  edge_logit_max<<<ceil_div((long long)Etot, 256), 256, 0, stream>>>(ei, E, N, 1, als, ald, segmax);
  edge_logit_sum<<<ceil_div((long long)Etot, 256), 256, 0, stream>>>(ei, E, N, 1, als, ald, segmax, denom);
  edge_agg_d40<<<ceil_div((long long)Etot * 32, 256), 256, 0, stream>>>(ei, E, N, als, ald, segmax, denom, buf_h, out);
  (void)out_size;
}